// KANGNN_two_29308856828308
// MI455X (gfx1250) — compile-verified
//
#include <hip/hip_runtime.h>
#include <hip/hip_bf16.h>

#define IN_F   64
#define MID_F  5
#define HID_F  128
#define NB     11      // G + K = 8 + 3 basis functions
#define NGR    512

typedef __attribute__((ext_vector_type(2))) float v2f;
typedef __attribute__((ext_vector_type(8))) float v8f;

// ---------------------------------------------------------------------------
// Uniform-knot cubic B-spline basis: knots[j] = -1.75 + 0.25*j, j = 0..14.
// Cox-de Boor, fully unrolled into registers. In-place forward update is
// exact: at depth d, b[j] consumes pre-update b[j], b[j+1].
// ---------------------------------------------------------------------------
__device__ __forceinline__ void bspline11(float x, float* __restrict__ out) {
    float b[14];
#pragma unroll
    for (int j = 0; j < 14; ++j) {
        float k0 = -1.75f + 0.25f * j;
        b[j] = (x >= k0 && x < k0 + 0.25f) ? 1.0f : 0.0f;
    }
#pragma unroll
    for (int d = 1; d <= 3; ++d) {
        float inv = 1.0f / (0.25f * d);
#pragma unroll
        for (int j = 0; j < 13; ++j) {
            if (j < 14 - d) {
                float kj    = -1.75f + 0.25f * j;
                float left  = (x - kj) * inv * b[j];
                float right = ((kj + 0.25f * (d + 1)) - x) * inv * b[j + 1];
                b[j] = left + right;
            }
        }
    }
#pragma unroll
    for (int j = 0; j < NB; ++j) out[j] = b[j];
}

__device__ __forceinline__ float silu(float x) { return x / (1.0f + __expf(-x)); }

// ---------------------------------------------------------------------------
// KAN encoder: [IN=64] -> [5] -> [HID=128], one 64-thread block per node.
// ---------------------------------------------------------------------------
__global__ void k_encoder(const float* __restrict__ h,
                          const float* __restrict__ coef1, const float* __restrict__ sb1,
                          const float* __restrict__ ss1,
                          const float* __restrict__ coef2, const float* __restrict__ sb2,
                          const float* __restrict__ ss2,
                          float* __restrict__ h0) {
    __shared__ float sB[IN_F][NB + 1];   // 11 basis + silu
    __shared__ float sMid[MID_F];
    __shared__ float sB2[MID_F][NB + 1];
    const int n = blockIdx.x;
    const int t = threadIdx.x;           // 64 threads

    float xi = h[(size_t)n * IN_F + t];
    float bas[NB];
    bspline11(xi, bas);
#pragma unroll
    for (int c = 0; c < NB; ++c) sB[t][c] = bas[c];
    sB[t][NB] = silu(xi);
    __syncthreads();

    if (t < MID_F) {
        float acc = 0.0f;
        for (int i = 0; i < IN_F; ++i) {
            float s = 0.0f;
#pragma unroll
            for (int c = 0; c < NB; ++c) s += coef1[(i * MID_F + t) * NB + c] * sB[i][c];
            acc += ss1[i * MID_F + t] * s + sb1[i * MID_F + t] * sB[i][NB];
        }
        sMid[t] = acc;
    }
    __syncthreads();

    if (t < MID_F) {
        float x2 = sMid[t];
        float b2[NB];
        bspline11(x2, b2);
#pragma unroll
        for (int c = 0; c < NB; ++c) sB2[t][c] = b2[c];
        sB2[t][NB] = silu(x2);
    }
    __syncthreads();

#pragma unroll
    for (int rep = 0; rep < 2; ++rep) {
        int o = t + rep * 64;
        float acc = 0.0f;
#pragma unroll
        for (int i = 0; i < MID_F; ++i) {
            float s = 0.0f;
#pragma unroll
            for (int c = 0; c < NB; ++c) s += coef2[(i * HID_F + o) * NB + c] * sB2[i][c];
            acc += ss2[i * HID_F + o] * s + sb2[i * HID_F + o] * sB2[i][NB];
        }
        h0[(size_t)n * HID_F + o] = acc;
    }
}

// ---------------------------------------------------------------------------
// CSC build step 1: in-degree histogram (int atomics, 1 per edge).
// ---------------------------------------------------------------------------
__global__ void k_count(const int* __restrict__ dst, int E, int* __restrict__ cnt) {
    int e = blockIdx.x * blockDim.x + threadIdx.x;
    if (e < E) atomicAdd(&cnt[dst[e]], 1);
}

// ---------------------------------------------------------------------------
// CSC build step 2: exclusive scan of cnt[0..N) -> rowstart[0..N].
// Single 1024-thread workgroup, chunked Hillis-Steele scan with LDS carry.
// ~100 chunks; negligible runtime, runs once per launch.
// ---------------------------------------------------------------------------
__global__ void k_scan(const int* __restrict__ cnt, int N, int* __restrict__ rowstart) {
    __shared__ int smem[1024];
    __shared__ int carry;
    const int t = threadIdx.x;
    if (t == 0) carry = 0;
    __syncthreads();
    for (int base = 0; base < N; base += 1024) {
        int v = (base + t < N) ? cnt[base + t] : 0;
        smem[t] = v;
        __syncthreads();
        for (int off = 1; off < 1024; off <<= 1) {
            int x = (t >= off) ? smem[t - off] : 0;
            __syncthreads();
            smem[t] += x;
            __syncthreads();
        }
        int excl = smem[t] - v;
        int c = carry;
        if (base + t < N) rowstart[base + t] = c + excl;
        __syncthreads();                       // everyone done reading carry/smem
        if (t == 0) carry += smem[1023];
        __syncthreads();                       // carry visible; smem reusable
    }
    if (t == 0) rowstart[N] = carry;
}

// ---------------------------------------------------------------------------
// CSC build step 3: bucket src ids by dst (1 cursor atomic per edge).
// ---------------------------------------------------------------------------
__global__ void k_fill(const int* __restrict__ src, const int* __restrict__ dst, int E,
                       const int* __restrict__ rowstart, int* __restrict__ cursor,
                       int* __restrict__ adjsrc) {
    int e = blockIdx.x * blockDim.x + threadIdx.x;
    if (e >= E) return;
    int d = dst[e];
    int pos = atomicAdd(&cursor[d], 1);
    adjsrc[rowstart[d] + pos] = src[e];
}

// ---------------------------------------------------------------------------
// Atomic-free mean aggregation: one wave32 per dst node. Lanes stream float4
// rows of h[src] (L2-resident: h is 51MB vs 192MB L2) and accumulate in regs;
// the mean (deg-normalized) is written directly.
// ---------------------------------------------------------------------------
__global__ void k_aggregate(const float* __restrict__ h, const int* __restrict__ rowstart,
                            const int* __restrict__ adjsrc, int N,
                            float* __restrict__ nmean) {
    int node = (blockIdx.x * blockDim.x + threadIdx.x) >> 5;
    int lane = threadIdx.x & 31;
    if (node >= N) return;
    int j0 = rowstart[node], j1 = rowstart[node + 1];
    float4 acc = make_float4(0.f, 0.f, 0.f, 0.f);
    for (int j = j0; j < j1; ++j) {
        int s = adjsrc[j];                      // uniform across the wave
        const float4 v = *reinterpret_cast<const float4*>(h + (size_t)s * HID_F + lane * 4);
        acc.x += v.x; acc.y += v.y; acc.z += v.z; acc.w += v.w;
    }
    float invd = 1.0f / fmaxf((float)(j1 - j0), 1.0f);
    float4 r = make_float4(acc.x * invd, acc.y * invd, acc.z * invd, acc.w * invd);
    *reinterpret_cast<float4*>(nmean + (size_t)node * HID_F + lane * 4) = r;
}

// ---------------------------------------------------------------------------
// SAGEConv: z = h@Ws + nmean@Wn + b + h, leaky_relu(0.01).
// 256 threads = 8 waves per block; block covers 16 nodes x 128 cols.
// Each wave owns a 16x16 f32 tile accumulated with V_WMMA_F32_16X16X4_F32.
// EXEC is all-ones throughout (N % 16 == 0, no divergence) per WMMA rules.
// ---------------------------------------------------------------------------
__global__ void k_sage(const float* __restrict__ hin, const float* __restrict__ nmean,
                       const float* __restrict__ Ws, const float* __restrict__ Wn,
                       const float* __restrict__ bias, float* __restrict__ hout) {
    __shared__ float sH[16][HID_F + 4];
    __shared__ float sM[16][HID_F + 4];
    const int n0 = blockIdx.x * 16;
    const int t  = threadIdx.x;                 // 256

    for (int idx = t; idx < 16 * HID_F; idx += 256) {
        int r = idx >> 7, c = idx & (HID_F - 1);
        int gr = n0 + r;
        sH[r][c] = hin[(size_t)gr * HID_F + c];
        sM[r][c] = nmean[(size_t)gr * HID_F + c];
    }
    __syncthreads();

    const int wv   = t >> 5;          // wave id -> column tile
    const int lane = t & 31;
    const int li   = lane & 15;
    const int half = lane >> 4;
    const int col  = wv * 16 + li;

    v8f acc = {};
#pragma unroll 4
    for (int k0 = 0; k0 < HID_F; k0 += 4) {
        const int kk = k0 + 2 * half;
        v2f ah; ah.x = sH[li][kk]; ah.y = sH[li][kk + 1];          // A: self features
        v2f am; am.x = sM[li][kk]; am.y = sM[li][kk + 1];          // A: mean neighbors
        v2f bs; bs.x = Ws[kk * HID_F + col]; bs.y = Ws[(kk + 1) * HID_F + col];
        v2f bn; bn.x = Wn[kk * HID_F + col]; bn.y = Wn[(kk + 1) * HID_F + col];
        acc = __builtin_amdgcn_wmma_f32_16x16x4_f32(false, ah, false, bs, (short)0, acc, false, false);
        acc = __builtin_amdgcn_wmma_f32_16x16x4_f32(false, am, false, bn, (short)0, acc, false, false);
    }

    const float bc = bias[col];
#pragma unroll
    for (int v = 0; v < 8; ++v) {
        int r  = v + 8 * half;                  // D layout: VGPR v -> rows v / v+8
        int gr = n0 + r;
        float z = acc[v] + bc + sH[r][col];     // + residual
        hout[(size_t)gr * HID_F + col] = (z > 0.0f) ? z : 0.01f * z;
    }
}

// ---------------------------------------------------------------------------
// Per-graph mean pool (atomic accumulate), one wave per node.
// ---------------------------------------------------------------------------
__global__ void k_pool(const float* __restrict__ h, const int* __restrict__ gid,
                       int N, float* __restrict__ pool, float* __restrict__ cnt) {
    int wave = (blockIdx.x * blockDim.x + threadIdx.x) >> 5;
    int lane = threadIdx.x & 31;
    if (wave >= N) return;
    int g = gid[wave];
    const float4 v = *reinterpret_cast<const float4*>(h + (size_t)wave * HID_F + lane * 4);
    float* p = pool + (size_t)g * HID_F + lane * 4;
    atomicAdd(p + 0, v.x);
    atomicAdd(p + 1, v.y);
    atomicAdd(p + 2, v.z);
    atomicAdd(p + 3, v.w);
    if (lane == 0) atomicAdd(&cnt[g], 1.0f);
}

// ---------------------------------------------------------------------------
// KAN readout [128]->[5]->[1] + sigmoid, one 128-thread block per graph.
// ---------------------------------------------------------------------------
__global__ void k_readout(const float* __restrict__ pool, const float* __restrict__ cnt,
                          const float* __restrict__ coefr1, const float* __restrict__ sbr1,
                          const float* __restrict__ ssr1,
                          const float* __restrict__ coefr2, const float* __restrict__ sbr2,
                          const float* __restrict__ ssr2,
                          float* __restrict__ out) {
    __shared__ float sB[HID_F][NB + 1];
    __shared__ float sMid[MID_F];
    __shared__ float sB2[MID_F][NB + 1];
    const int g = blockIdx.x;
    const int t = threadIdx.x;              // 128

    float inv = 1.0f / fmaxf(cnt[g], 1.0f);
    float x = pool[(size_t)g * HID_F + t] * inv;
    float bas[NB];
    bspline11(x, bas);
#pragma unroll
    for (int c = 0; c < NB; ++c) sB[t][c] = bas[c];
    sB[t][NB] = silu(x);
    __syncthreads();

    if (t < MID_F) {
        float acc = 0.0f;
        for (int i = 0; i < HID_F; ++i) {
            float s = 0.0f;
#pragma unroll
            for (int c = 0; c < NB; ++c) s += coefr1[(i * MID_F + t) * NB + c] * sB[i][c];
            acc += ssr1[i * MID_F + t] * s + sbr1[i * MID_F + t] * sB[i][NB];
        }
        sMid[t] = acc;
    }
    __syncthreads();

    if (t < MID_F) {
        float x2 = sMid[t];
        float b2[NB];
        bspline11(x2, b2);
#pragma unroll
        for (int c = 0; c < NB; ++c) sB2[t][c] = b2[c];
        sB2[t][NB] = silu(x2);
    }
    __syncthreads();

    if (t == 0) {
        float acc = 0.0f;
#pragma unroll
        for (int i = 0; i < MID_F; ++i) {
            float s = 0.0f;
#pragma unroll
            for (int c = 0; c < NB; ++c) s += coefr2[i * NB + c] * sB2[i][c];
            acc += ssr2[i] * s + sbr2[i] * sB2[i][NB];
        }
        out[g] = 1.0f / (1.0f + __expf(-acc));
    }
}

// ---------------------------------------------------------------------------
extern "C" void kernel_launch(void* const* d_in, const int* in_sizes, int n_in,
                              void* d_out, int out_size, void* d_ws, size_t ws_size,
                              hipStream_t stream) {
    const float* h     = (const float*)d_in[0];
    const int*   src   = (const int*)d_in[1];
    const int*   dst   = (const int*)d_in[2];
    const int*   gid   = (const int*)d_in[3];
    const float* coef1 = (const float*)d_in[4];
    const float* sb1   = (const float*)d_in[5];
    const float* ss1   = (const float*)d_in[6];
    const float* coef2 = (const float*)d_in[7];
    const float* sb2   = (const float*)d_in[8];
    const float* ss2   = (const float*)d_in[9];
    const float* Ws0   = (const float*)d_in[10];
    const float* Wn0   = (const float*)d_in[11];
    const float* b0    = (const float*)d_in[12];
    const float* Ws1   = (const float*)d_in[13];
    const float* Wn1   = (const float*)d_in[14];
    const float* b1    = (const float*)d_in[15];
    const float* coefr1= (const float*)d_in[16];
    const float* sbr1  = (const float*)d_in[17];
    const float* ssr1  = (const float*)d_in[18];
    const float* coefr2= (const float*)d_in[19];
    const float* sbr2  = (const float*)d_in[20];
    const float* ssr2  = (const float*)d_in[21];

    const int N = in_sizes[0] / IN_F;     // 100000 (divisible by 16)
    const int E = in_sizes[1];            // 1600000

    // ---- workspace layout -------------------------------------------------
    float* fws   = (float*)d_ws;
    float* hA    = fws;                                  // N*128
    float* hB    = hA + (size_t)N * HID_F;               // N*128
    float* nmean = hB + (size_t)N * HID_F;               // N*128
    float* pool  = nmean + (size_t)N * HID_F;            // 512*128
    float* cntg  = pool + (size_t)NGR * HID_F;           // 512
    int*   iws      = (int*)(cntg + NGR);
    int*   cnt      = iws;                               // N
    int*   rowstart = cnt + N;                           // N+1
    int*   cursor   = rowstart + (N + 1);                // N
    int*   adjsrc   = cursor + N;                        // E

    // ---- zero accumulators (capture-safe async memsets) -------------------
    hipMemsetAsync(cnt, 0, (size_t)N * sizeof(int), stream);
    hipMemsetAsync(cursor, 0, (size_t)N * sizeof(int), stream);
    hipMemsetAsync(pool, 0, ((size_t)NGR * HID_F + NGR) * sizeof(float), stream);

    // ---- KAN encoder ------------------------------------------------------
    k_encoder<<<N, 64, 0, stream>>>(h, coef1, sb1, ss1, coef2, sb2, ss2, hA);

    // ---- build CSC adjacency (once per launch) ----------------------------
    k_count<<<(E + 255) / 256, 256, 0, stream>>>(dst, E, cnt);
    k_scan<<<1, 1024, 0, stream>>>(cnt, N, rowstart);
    k_fill<<<(E + 255) / 256, 256, 0, stream>>>(src, dst, E, rowstart, cursor, adjsrc);

    // ---- SAGE layer 0 -----------------------------------------------------
    k_aggregate<<<(N + 7) / 8, 256, 0, stream>>>(hA, rowstart, adjsrc, N, nmean);
    k_sage<<<N / 16, 256, 0, stream>>>(hA, nmean, Ws0, Wn0, b0, hB);

    // ---- SAGE layer 1 -----------------------------------------------------
    k_aggregate<<<(N + 7) / 8, 256, 0, stream>>>(hB, rowstart, adjsrc, N, nmean);
    k_sage<<<N / 16, 256, 0, stream>>>(hB, nmean, Ws1, Wn1, b1, hA);

    // ---- pool + readout ---------------------------------------------------
    k_pool<<<(N + 7) / 8, 256, 0, stream>>>(hA, gid, N, pool, cntg);
    k_readout<<<NGR, 128, 0, stream>>>(pool, cntg, coefr1, sbr1, ssr1,
                                       coefr2, sbr2, ssr2, (float*)d_out);
}